// ModelBaseline_4449586118670
// MI455X (gfx1250) — compile-verified
//
#include <hip/hip_runtime.h>
#include <cstdint>

#define Nn   100000
#define Ee   100000
#define Mm   1600000
#define HIDc 128
#define OUTc 64
#define Lc   2
#define LDA  132   // padded LDS row stride (floats) to avoid bank conflicts

typedef float v2f __attribute__((ext_vector_type(2)));
typedef float v8f __attribute__((ext_vector_type(8)));

__device__ __forceinline__ float lrelu(float v) { return v > 0.0f ? v : 0.01f * v; }

// -------- fused (optional LayerNorm) + GEMM via V_WMMA_F32_16X16X4_F32 --------
// X: [rows, 128] row-major.  W: [128, OUTC] row-major.  Y: [rows, OUTC].
// Block: (OUTC/16) waves; each wave produces one 16x16 tile of Y, K accumulated
// in steps of 4 with f32 WMMA.  16-row A tile staged (and LN'd) in LDS.
template <int OUTC, bool DO_LN, bool DO_RELU>
__global__ __launch_bounds__((OUTC / 16) * 32)
void ln_gemm_kernel(const float* __restrict__ X, const float* __restrict__ g,
                    const float* __restrict__ bln, const float* __restrict__ W,
                    const float* __restrict__ bias, float* __restrict__ Y) {
  __shared__ float A[16 * LDA];
  __shared__ float sM[16], sR[16];
  const int tid = threadIdx.x;
  constexpr int NTHR = (OUTC / 16) * 32;
  const int rowbase = blockIdx.x * 16;

  // stage 16x128 input tile into LDS (float4)
  for (int i = tid * 4; i < 16 * HIDc; i += NTHR * 4) {
    const int r = i >> 7, c = i & 127;
    const float4 v = *(const float4*)&X[(size_t)(rowbase + r) * HIDc + c];
    *(float4*)&A[r * LDA + c] = v;
  }
  __syncthreads();

  if (DO_LN) {
    if (tid < 16) {
      float s = 0.f, ss = 0.f;
      for (int c = 0; c < HIDc; ++c) { const float v = A[tid * LDA + c]; s += v; ss += v * v; }
      const float m = s * (1.0f / HIDc);
      const float var = ss * (1.0f / HIDc) - m * m;
      sM[tid] = m;
      sR[tid] = rsqrtf(var + 1e-5f);
    }
    __syncthreads();
    for (int i = tid; i < 16 * HIDc; i += NTHR) {
      const int r = i >> 7, c = i & 127;
      A[r * LDA + c] = (A[r * LDA + c] - sM[r]) * sR[r] * g[c] + bln[c];
    }
    __syncthreads();
  }

  // ---- WMMA phase: D(16x16) += A(16x4) * B(4x16), K = 0..124 step 4 ----
  const int lane = tid & 31;
  const int wave = tid >> 5;
  const int half = lane >> 4;   // 0: lanes 0-15, 1: lanes 16-31
  const int lrow = lane & 15;   // M index for A frag / N index for B,C,D frags
  const int col  = wave * 16 + lrow;

  v8f acc = {0.f, 0.f, 0.f, 0.f, 0.f, 0.f, 0.f, 0.f};
  for (int k0 = 0; k0 < HIDc; k0 += 4) {
    // A frag (ISA 16x4 f32 layout): lanes 0-15 hold K=k0,k0+1; lanes 16-31 K=k0+2,k0+3
    const v2f a = *(const v2f*)&A[lrow * LDA + k0 + 2 * half];
    v2f b;
    b[0] = W[(size_t)(k0 + 2 * half)     * OUTC + col];
    b[1] = W[(size_t)(k0 + 2 * half + 1) * OUTC + col];
    acc = __builtin_amdgcn_wmma_f32_16x16x4_f32(false, a, false, b, (short)0, acc,
                                                false, false);
  }

  const float bb = bias ? bias[col] : 0.0f;
#pragma unroll
  for (int r = 0; r < 8; ++r) {  // C/D layout: VGPR r -> row r (half 0) / row 8+r (half 1)
    float v = acc[r] + bb;
    if (DO_RELU) v = lrelu(v);
    Y[(size_t)(rowbase + half * 8 + r) * OUTC + col] = v;
  }
}

// -------- degree counting + reciprocal --------
__global__ void degree_kernel(const int* __restrict__ src, const int* __restrict__ dst,
                              float* __restrict__ degD, float* __restrict__ degB) {
  const int m = blockIdx.x * blockDim.x + threadIdx.x;
  if (m < Mm) {
    atomicAdd(&degD[src[m]], 1.0f);
    atomicAdd(&degB[dst[m]], 1.0f);
  }
}

__global__ void recip_kernel(float* __restrict__ d, int n) {
  const int i = blockIdx.x * blockDim.x + threadIdx.x;
  if (i < n) d[i] = d[i] > 0.0f ? 1.0f / d[i] : 0.0f;
}

// -------- segment-sum scatter: out[out_idx[m]] += scale[in_idx[m]] * xin[in_idx[m]] --------
// one wave per incidence; each lane handles 4 consecutive channels (float4 gather,
// 4 scalar global_atomic_add_f32)
__global__ void scatter_add_kernel(const int* __restrict__ out_idx,
                                   const int* __restrict__ in_idx,
                                   const float* __restrict__ xin,
                                   float* __restrict__ out,
                                   const float* __restrict__ scale) {
  const long gid = (long)blockIdx.x * blockDim.x + threadIdx.x;
  const int m = (int)(gid >> 5);
  if (m >= Mm) return;
  const int c = ((int)gid & 31) << 2;
  const int si = in_idx[m];
  const int di = out_idx[m];
  const float s = scale ? scale[si] : 1.0f;
  const float4 v = *(const float4*)&xin[(size_t)si * HIDc + c];
  float* o = &out[(size_t)di * HIDc + c];
  atomicAdd(o + 0, s * v.x);
  atomicAdd(o + 1, s * v.y);
  atomicAdd(o + 2, s * v.z);
  atomicAdd(o + 3, s * v.w);
}

// -------- h = leaky_relu(D[n] * h2_raw + conv_b) --------
__global__ void finalize_kernel(const float* __restrict__ h2raw, const float* __restrict__ Dv,
                                const float* __restrict__ cb, float* __restrict__ h) {
  const int i = blockIdx.x * blockDim.x + threadIdx.x;
  if (i < Nn * HIDc) {
    const int n = i >> 7, c = i & 127;
    h[i] = lrelu(Dv[n] * h2raw[i] + cb[c]);
  }
}

// -------- segment-min via order-preserving uint transform --------
__device__ __forceinline__ unsigned flipf(float f) {
  const unsigned u = __float_as_uint(f);
  return (u & 0x80000000u) ? ~u : (u | 0x80000000u);
}

__global__ void fill_kernel(unsigned* __restrict__ p, unsigned val, int n) {
  const int i = blockIdx.x * blockDim.x + threadIdx.x;
  if (i < n) p[i] = val;
}

__global__ void scatter_min_kernel(const int* __restrict__ src, const int* __restrict__ dst,
                                   const float* __restrict__ h, unsigned* __restrict__ agg) {
  const long gid = (long)blockIdx.x * blockDim.x + threadIdx.x;
  const int m = (int)(gid >> 5);
  if (m >= Mm) return;
  const int c = ((int)gid & 31) << 2;
  const int si = src[m];
  const int di = dst[m];
  const float4 v = *(const float4*)&h[(size_t)si * HIDc + c];
  unsigned* o = &agg[(size_t)di * HIDc + c];
  atomicMin(o + 0, flipf(v.x));
  atomicMin(o + 1, flipf(v.y));
  atomicMin(o + 2, flipf(v.z));
  atomicMin(o + 3, flipf(v.w));
}

__global__ void unflip_kernel(unsigned* __restrict__ p, int n) {
  const int i = blockIdx.x * blockDim.x + threadIdx.x;
  if (i < n) {
    const unsigned u = p[i];
    p[i] = (u & 0x80000000u) ? (u ^ 0x80000000u) : ~u;
  }
}

extern "C" void kernel_launch(void* const* d_in, const int* in_sizes, int n_in,
                              void* d_out, int out_size, void* d_ws, size_t ws_size,
                              hipStream_t stream) {
  (void)in_sizes; (void)n_in; (void)out_size; (void)ws_size;

  const float* x    = (const float*)d_in[0];
  // d_in[1] (x_e) unused by the reference computation
  const int*   edge = (const int*)d_in[2];
  const int*   src  = edge;         // edge_index[0]
  const int*   dst  = edge + Mm;    // edge_index[1]
  const float* in_g = (const float*)d_in[3];
  const float* in_b = (const float*)d_in[4];
  const float* W0   = (const float*)d_in[5];
  const float* b0   = (const float*)d_in[6];
  const float* ng   = (const float*)d_in[7];
  const float* nb   = (const float*)d_in[8];
  const float* cW   = (const float*)d_in[9];
  const float* cb   = (const float*)d_in[10];
  const float* lW   = (const float*)d_in[11];
  const float* lb   = (const float*)d_in[12];
  float* out = (float*)d_out;

  // workspace layout (fp32): h[N*128] | xw/h2_raw[N*128] | he_raw/agg[E*128] | D[N] | B[E]
  float* h  = (float*)d_ws;
  float* xw = h  + (size_t)Nn * HIDc;
  float* he = xw + (size_t)Nn * HIDc;
  float* Dv = he + (size_t)Ee * HIDc;
  float* Bv = Dv + Nn;

  // degrees -> reciprocals
  hipMemsetAsync(Dv, 0, sizeof(float) * (Nn + Ee), stream);
  degree_kernel<<<(Mm + 255) / 256, 256, 0, stream>>>(src, dst, Dv, Bv);
  recip_kernel<<<(Nn + Ee + 255) / 256, 256, 0, stream>>>(Dv, Nn + Ee);

  // h = leaky_relu(LN(x) @ in_proj_W + in_proj_b)
  ln_gemm_kernel<HIDc, true, true><<<Nn / 16, 256, 0, stream>>>(x, in_g, in_b, W0, b0, h);

  const int sc_blocks = (int)(((long)Mm * 32) / 256);  // one wave per incidence
  for (int i = 0; i < Lc; ++i) {
    // xw = LN(h) @ conv_W[i]
    ln_gemm_kernel<HIDc, true, false><<<Nn / 16, 256, 0, stream>>>(
        h, ng + i * HIDc, nb + i * HIDc, cW + (size_t)i * HIDc * HIDc, nullptr, xw);
    // he_raw[e] = sum_{m: dst=e} xw[src[m]]
    hipMemsetAsync(he, 0, sizeof(float) * (size_t)Ee * HIDc, stream);
    scatter_add_kernel<<<sc_blocks, 256, 0, stream>>>(dst, src, xw, he, nullptr);
    // h2_raw[n] = sum_{m: src=n} B[dst[m]] * he_raw[dst[m]]   (reuse xw buffer)
    hipMemsetAsync(xw, 0, sizeof(float) * (size_t)Nn * HIDc, stream);
    scatter_add_kernel<<<sc_blocks, 256, 0, stream>>>(src, dst, he, xw, Bv);
    // h = leaky_relu(D * h2_raw + conv_b[i])
    finalize_kernel<<<(Nn * HIDc + 255) / 256, 256, 0, stream>>>(xw, Dv, cb + i * HIDc, h);
  }

  // agg = segment_min(h[src], dst)   (reuse he buffer as uint)
  unsigned* agg = (unsigned*)he;
  fill_kernel<<<(Ee * HIDc + 255) / 256, 256, 0, stream>>>(agg, 0xFF800000u /*flip(+inf)*/,
                                                           Ee * HIDc);
  scatter_min_kernel<<<sc_blocks, 256, 0, stream>>>(src, dst, h, agg);
  unflip_kernel<<<(Ee * HIDc + 255) / 256, 256, 0, stream>>>(agg, Ee * HIDc);

  // out = agg @ lin_W + lin_b
  ln_gemm_kernel<OUTc, false, false><<<Ee / 16, 128, 0, stream>>>(
      (const float*)agg, nullptr, nullptr, lW, lb, out);
}